// TriangleCollisionLoss_16166256902861
// MI455X (gfx1250) — compile-verified
//
#include <hip/hip_runtime.h>
#include <hip/hip_bf16.h>

typedef __attribute__((ext_vector_type(2))) float v2f;
typedef __attribute__((ext_vector_type(8))) float v8f;

#define NFACES 4096
#define NTILES 256          /* NFACES / 16 */
#define JSTRIP 32           /* j-tiles per wave */
#define NRM_EPS 1e-12f

// ---------------------------------------------------------------------------
// Kernel 1: per-face preprocessing.  Builds WMMA operand arrays:
//   AN[f]  = (nx, ny, nz, -refdot)          A operand for plane-distance WMMA
//   AC[f]  = (-2cx, -2cy, -2cz, |c|^2)      A operand for distance WMMA
//   BC[f]  = (cx, cy, cz, 1)                B operand for distance WMMA
//   BV[k][f] = (vx, vy, vz, 1)              B operands (k=0,1,2) for plane WMMA
//   CN2[f] = |c|^2                          per-column threshold source
//   CNT[f] = 0                              collision-count accumulator
// ---------------------------------------------------------------------------
__global__ void __launch_bounds__(256)
tri_prep_kernel(const float* __restrict__ verts,
                const int*   __restrict__ faces,
                float* __restrict__ AN, float* __restrict__ AC,
                float* __restrict__ BC, float* __restrict__ BV,
                float* __restrict__ CN2, float* __restrict__ CNT) {
    int f = blockIdx.x * blockDim.x + threadIdx.x;
    if (f >= NFACES) return;

    int i0 = faces[f * 3 + 0];
    int i1 = faces[f * 3 + 1];
    int i2 = faces[f * 3 + 2];

    float p0x = verts[i0 * 3 + 0], p0y = verts[i0 * 3 + 1], p0z = verts[i0 * 3 + 2];
    float p1x = verts[i1 * 3 + 0], p1y = verts[i1 * 3 + 1], p1z = verts[i1 * 3 + 2];
    float p2x = verts[i2 * 3 + 0], p2y = verts[i2 * 3 + 1], p2z = verts[i2 * 3 + 2];

    float ax = p1x - p0x, ay = p1y - p0y, az = p1z - p0z;
    float bx = p2x - p0x, by = p2y - p0y, bz = p2z - p0z;

    // cross product -> face normal, normalized with torch-style eps clamp
    float nx = ay * bz - az * by;
    float ny = az * bx - ax * bz;
    float nz = ax * by - ay * bx;
    float nrm = sqrtf(nx * nx + ny * ny + nz * nz);
    float inv = 1.0f / fmaxf(nrm, NRM_EPS);
    nx *= inv; ny *= inv; nz *= inv;

    float cx = (p0x + p1x + p2x) * (1.0f / 3.0f);
    float cy = (p0y + p1y + p2y) * (1.0f / 3.0f);
    float cz = (p0z + p1z + p2z) * (1.0f / 3.0f);
    float cc = cx * cx + cy * cy + cz * cz;

    float refdot = p0x * nx + p0y * ny + p0z * nz;

    AN[f * 4 + 0] = nx;  AN[f * 4 + 1] = ny;  AN[f * 4 + 2] = nz;  AN[f * 4 + 3] = -refdot;
    AC[f * 4 + 0] = -2.0f * cx; AC[f * 4 + 1] = -2.0f * cy;
    AC[f * 4 + 2] = -2.0f * cz; AC[f * 4 + 3] = cc;
    BC[f * 4 + 0] = cx;  BC[f * 4 + 1] = cy;  BC[f * 4 + 2] = cz;  BC[f * 4 + 3] = 1.0f;

    BV[0 * 4 * NFACES + f * 4 + 0] = p0x; BV[0 * 4 * NFACES + f * 4 + 1] = p0y;
    BV[0 * 4 * NFACES + f * 4 + 2] = p0z; BV[0 * 4 * NFACES + f * 4 + 3] = 1.0f;
    BV[1 * 4 * NFACES + f * 4 + 0] = p1x; BV[1 * 4 * NFACES + f * 4 + 1] = p1y;
    BV[1 * 4 * NFACES + f * 4 + 2] = p1z; BV[1 * 4 * NFACES + f * 4 + 3] = 1.0f;
    BV[2 * 4 * NFACES + f * 4 + 0] = p2x; BV[2 * 4 * NFACES + f * 4 + 1] = p2y;
    BV[2 * 4 * NFACES + f * 4 + 2] = p2z; BV[2 * 4 * NFACES + f * 4 + 3] = 1.0f;

    CN2[f] = cc;
    CNT[f] = 0.0f;
}

// ---------------------------------------------------------------------------
// Kernel 2: one wave per (i-tile, 32-tile j-strip).  A operands stay in 4
// VGPRs across the strip.  Loop body: 5 loads + 4 K=4 f32 WMMAs (all with
// inline C=0) + sign tests.  The |c_j|^2 term of the squared distance is
// folded into the comparison threshold (d2 < 1 - |c_j|^2) instead of a
// broadcast C matrix, so the d2 WMMA doesn't wait on the CN2 load.  Diagonal
// exclusion reuses the stored hit indicator under a wave-uniform branch.
// Grid has no partial waves -> EXEC all-ones at every WMMA.
// ---------------------------------------------------------------------------
__global__ void __launch_bounds__(256)
tri_pairs_kernel(const float* __restrict__ AN, const float* __restrict__ AC,
                 const float* __restrict__ BC, const float* __restrict__ BV,
                 const float* __restrict__ CN2, float* __restrict__ CNT) {
    int wid  = blockIdx.x * 8 + (threadIdx.x >> 5);   // 2048 waves
    int lane = threadIdx.x & 31;
    int ti   = wid >> 3;            // i-tile: 0..255
    int jch  = wid & 7;             // j-strip: 0..7 (32 tiles each)
    int m    = lane & 15;           // A: row M    B: column N
    int half = lane >> 4;           // K pair {0,1} vs {2,3}; D rows M vs M+8

    // A 16x4 f32 layout: lane m holds row M=m; VGPR0/1 = K = 2*half, 2*half+1
    const int iofs = (ti * 16 + m) * 4 + half * 2;
    const v2f a_n = *(const v2f*)(AN + iofs);
    const v2f a_c = *(const v2f*)(AC + iofs);

    float acc[8] = {0.f, 0.f, 0.f, 0.f, 0.f, 0.f, 0.f, 0.f};
    const v8f zero = {0.f, 0.f, 0.f, 0.f, 0.f, 0.f, 0.f, 0.f};

    const int jt0 = jch * JSTRIP;
    for (int t = 0; t < JSTRIP; ++t) {
        const int tj   = jt0 + t;
        const int jofs = (tj * 16 + m) * 4 + half * 2;

        // B 4x16 f32 layout mirrors A: lane m holds column N=m.
        v2f b_c = *(const v2f*)(BC + jofs);
        v2f b_0 = *(const v2f*)(BV + 0 * 4 * NFACES + jofs);
        v2f b_1 = *(const v2f*)(BV + 1 * 4 * NFACES + jofs);
        v2f b_2 = *(const v2f*)(BV + 2 * 4 * NFACES + jofs);

        // nearby threshold: |c_i|^2 - 2 c_i.c_j < 1 - |c_j|^2
        float thr = 1.0f - CN2[tj * 16 + m];

        // d2p[i][j] = |c_i|^2 - 2 c_i.c_j           (C = inline 0)
        v8f d2p = __builtin_amdgcn_wmma_f32_16x16x4_f32(
            false, a_c, false, b_c, (short)0, zero, false, false);
        // S_k[i][j] = tri[j][k] . n_i - refdot_i    (C = inline 0)
        v8f s0 = __builtin_amdgcn_wmma_f32_16x16x4_f32(
            false, a_n, false, b_0, (short)0, zero, false, false);
        v8f s1 = __builtin_amdgcn_wmma_f32_16x16x4_f32(
            false, a_n, false, b_1, (short)0, zero, false, false);
        v8f s2 = __builtin_amdgcn_wmma_f32_16x16x4_f32(
            false, a_n, false, b_2, (short)0, zero, false, false);

        // roll(S,1,axis=2): products S0*S2, S1*S0, S2*S1; hit if any negative
        float h[8];
#pragma unroll
        for (int r = 0; r < 8; ++r) {
            bool hit = (d2p[r] < thr) &
                       ((s0[r] * s2[r] < 0.0f) |
                        (s1[r] * s0[r] < 0.0f) |
                        (s2[r] * s1[r] < 0.0f));
            h[r] = hit ? 1.0f : 0.0f;
            acc[r] += h[r];
        }

        // Diagonal exclusion: only the tile with tj == ti holds i == j
        // elements (one per accumulator row: column m == row r + 8*half).
        if (tj == ti) {
#pragma unroll
            for (int r = 0; r < 8; ++r)
                acc[r] -= (m == r + half * 8) ? h[r] : 0.0f;
        }
    }

    // Butterfly-reduce over the 16 lanes (columns) of each half.
#pragma unroll
    for (int r = 0; r < 8; ++r) {
        float v = acc[r];
        v += __shfl_xor(v, 1, 32);
        v += __shfl_xor(v, 2, 32);
        v += __shfl_xor(v, 4, 32);
        v += __shfl_xor(v, 8, 32);
        acc[r] = v;
    }

    if (m == 0) {
#pragma unroll
        for (int r = 0; r < 8; ++r)
            atomicAdd(&CNT[ti * 16 + r + half * 8], acc[r]);
    }
}

// ---------------------------------------------------------------------------
// Kernel 3: loss = mean(face_probs * collision_count)
// ---------------------------------------------------------------------------
__global__ void __launch_bounds__(256)
tri_reduce_kernel(const float* __restrict__ probs,
                  const float* __restrict__ CNT,
                  float* __restrict__ out) {
    __shared__ float sh[256];
    float s = 0.0f;
    for (int f = threadIdx.x; f < NFACES; f += 256)
        s += probs[f] * CNT[f];
    sh[threadIdx.x] = s;
    __syncthreads();
    for (int off = 128; off > 0; off >>= 1) {
        if (threadIdx.x < (unsigned)off) sh[threadIdx.x] += sh[threadIdx.x + off];
        __syncthreads();
    }
    if (threadIdx.x == 0) out[0] = sh[0] * (1.0f / (float)NFACES);
}

extern "C" void kernel_launch(void* const* d_in, const int* in_sizes, int n_in,
                              void* d_out, int out_size, void* d_ws, size_t ws_size,
                              hipStream_t stream) {
    const float* verts = (const float*)d_in[0];   // [V,3] f32
    const int*   faces = (const int*)d_in[1];     // [F,3] int
    const float* probs = (const float*)d_in[2];   // [F]   f32

    float* ws  = (float*)d_ws;
    float* AN  = ws;                      // 4*F
    float* AC  = AN  + 4 * NFACES;        // 4*F
    float* BC  = AC  + 4 * NFACES;        // 4*F
    float* BV  = BC  + 4 * NFACES;        // 12*F
    float* CN2 = BV  + 12 * NFACES;       // F
    float* CNT = CN2 + NFACES;            // F

    tri_prep_kernel<<<NFACES / 256, 256, 0, stream>>>(verts, faces, AN, AC, BC, BV, CN2, CNT);

    // 2048 waves: 256 i-tiles x 8 j-strips, 8 waves per block -> 256 blocks.
    // Every wave fully populated (EXEC all-ones, required by WMMA).
    tri_pairs_kernel<<<(NTILES * (NTILES / JSTRIP)) / 8, 256, 0, stream>>>(AN, AC, BC, BV, CN2, CNT);

    tri_reduce_kernel<<<1, 256, 0, stream>>>(probs, CNT, (float*)d_out);
}